// CIFModule_39822936768577
// MI455X (gfx1250) — compile-verified
//
#include <hip/hip_runtime.h>

#define NB 16
#define NT 4096
#define ND 512

typedef float v2f __attribute__((ext_vector_type(2)));
typedef float v8f __attribute__((ext_vector_type(8)));

// ---- workspace layout (in floats/ints) ----
#define OFF_CGP   0          // sum(gamma*proj_w)
#define OFF_CBP   1          // sum(beta*proj_w)
#define OFF_SUMA  16         // [NB] sum_alpha
#define OFF_SCALE 32         // [NB] rescale factor
#define OFF_FIDX  48         // [NB] final fidx (int)
#define OFF_TAIL  64         // [NB] tail flag (int)
#define OFF_CUTF  80         // [NB*17] fire-count at cuts (int)
#define OFF_CUTT  352        // [NB*17] last-fire time at cuts (int)
#define OFF_SA    1024                    // [NB*NT] a_t
#define OFF_SREM  (OFF_SA + NB*NT)        // [NB*NT] rem
#define OFF_SOVF  (OFF_SREM + NB*NT)      // [NB*NT] ovf
#define OFF_SPOS  (OFF_SOVF + NB*NT)      // [NB*NT] pos or -1 (int)

// ------------------------------------------------------------------
__global__ void k_zero(float* __restrict__ p, long n) {
  long i = ((long)blockIdx.x * blockDim.x + threadIdx.x) * 4;
  if (i + 3 < n) {
    *(float4*)(p + i) = make_float4(0.f, 0.f, 0.f, 0.f);
  } else {
    for (long k = i; k < n; ++k) p[k] = 0.f;
  }
}

// ------------------------------------------------------------------
__global__ void k_pre(const float* __restrict__ gamma, const float* __restrict__ beta,
                      const float* __restrict__ pw, float* __restrict__ wsf) {
  __shared__ float s1[256], s2[256];
  int tid = threadIdx.x;
  float a = 0.f, b2 = 0.f;
  for (int d = tid; d < ND; d += 256) { a += gamma[d] * pw[d]; b2 += beta[d] * pw[d]; }
  s1[tid] = a; s2[tid] = b2; __syncthreads();
  for (int o = 128; o > 0; o >>= 1) {
    if (tid < o) { s1[tid] += s1[tid + o]; s2[tid] += s2[tid + o]; }
    __syncthreads();
  }
  if (tid == 0) { wsf[OFF_CGP] = s1[0]; wsf[OFF_CBP] = s2[0]; }
}

// ------------------------------------------------------------------
// One wave per 16-timestep tile. Stage 18 enc rows (halo for conv3) into
// bank-padded LDS, then run K=512 in chunks of 4 through two
// V_WMMA_F32_16X16X4_F32: A = [ones; gamma*proj_w chunk], B = conv outputs
// (and their squares) for 16 t's. D rows 0/1 give Sum_y, Sum_{y*g*pw}
// (resp. Sum_{y^2}) for all 16 timesteps at once.
__global__ __launch_bounds__(32) void k_alpha(
    const float* __restrict__ enc, const int* __restrict__ lens,
    const float* __restrict__ convw, const float* __restrict__ gamma,
    const float* __restrict__ projw, const float* __restrict__ wsf,
    const float* __restrict__ projb, float* __restrict__ alpha) {
  __shared__ __align__(16) float sm[18 * 516 + 4 * ND];  // tile + w0/w1/w2 + pwg
  const int lane = threadIdx.x;
  const int b = blockIdx.x >> 8;
  const int t0 = (blockIdx.x & 255) << 4;
  const float* eb = enc + (long)b * NT * ND;

  // stage 18 rows (t0-1 .. t0+16) with zero padding, row stride 516 words
  for (int s = lane; s < 18 * 128; s += 32) {
    int r = s >> 7, col = s & 127;
    int tr = t0 - 1 + r;
    float4 v = make_float4(0.f, 0.f, 0.f, 0.f);
    if (tr >= 0 && tr < NT) v = *(const float4*)(eb + (long)tr * ND + col * 4);
    *(float4*)(sm + r * 516 + col * 4) = v;
  }
  float* w0 = sm + 18 * 516;
  float* w1 = w0 + ND;
  float* w2 = w1 + ND;
  float* pg = w2 + ND;
  for (int d = lane; d < ND; d += 32) {
    w0[d] = convw[d * 3 + 0];
    w1[d] = convw[d * 3 + 1];
    w2[d] = convw[d * 3 + 2];
    pg[d] = gamma[d] * projw[d];
  }
  __syncthreads();

  const int half = lane >> 4, tt = lane & 15;
  v8f c1 = {};  // row0: sum y ; row1: sum y*g*pw
  v8f c2 = {};  // row0: sum y^2
  for (int c = 0; c < ND; c += 4) {
    const int db = c + 2 * half;
    v2f em = *(const v2f*)(sm + tt * 516 + db);
    v2f ec = *(const v2f*)(sm + (tt + 1) * 516 + db);
    v2f ep = *(const v2f*)(sm + (tt + 2) * 516 + db);
    v2f a0 = *(const v2f*)(w0 + db);
    v2f a1 = *(const v2f*)(w1 + db);
    v2f a2 = *(const v2f*)(w2 + db);
    v2f y, y2, A;
    y.x = em.x * a0.x + ec.x * a1.x + ep.x * a2.x;
    y.y = em.y * a0.y + ec.y * a1.y + ep.y * a2.y;
    y2.x = y.x * y.x; y2.y = y.y * y.y;
    v2f pw = *(const v2f*)(pg + db);
    A.x = (tt == 0) ? 1.0f : ((tt == 1) ? pw.x : 0.0f);
    A.y = (tt == 0) ? 1.0f : ((tt == 1) ? pw.y : 0.0f);
    c1 = __builtin_amdgcn_wmma_f32_16x16x4_f32(false, A, false, y, (short)0, c1, false, false);
    c2 = __builtin_amdgcn_wmma_f32_16x16x4_f32(false, A, false, y2, (short)0, c2, false, false);
  }

  if (lane < 16) {
    float Sy = c1[0], Spw = c1[1], Syy = c2[0];
    float mu = Sy * (1.0f / ND);
    float var = Syy * (1.0f / ND) - mu * mu;
    float inv = rsqrtf(var + 1e-5f);
    float dot = inv * (Spw - mu * wsf[OFF_CGP]) + wsf[OFF_CBP] + projb[0];
    float al = 1.0f / (1.0f + expf(-dot));
    int t = t0 + lane;
    if (t >= lens[b]) al = 0.0f;
    alpha[(long)b * NT + t] = al;
  }
}

// ------------------------------------------------------------------
__global__ void k_sum(const float* __restrict__ alpha, float* __restrict__ wsf) {
  __shared__ float red[256];
  int b = blockIdx.x, tid = threadIdx.x;
  const float* ap = alpha + (long)b * NT;
  float s = 0.f;
  for (int t = tid; t < NT; t += 256) s += ap[t];
  red[tid] = s; __syncthreads();
  for (int o = 128; o > 0; o >>= 1) {
    if (tid < o) red[tid] += red[tid + o];
    __syncthreads();
  }
  if (tid == 0) wsf[OFF_SUMA + b] = red[0];
}

// ------------------------------------------------------------------
// Scalar CIF scan: 16 lanes, one batch each. Records per-t schedule and
// fire-aligned cut points so the heavy kernel parallelizes over 16 intervals.
__global__ __launch_bounds__(32) void k_sched(
    const float* __restrict__ alpha, const int* __restrict__ lens,
    const int* __restrict__ tgt, float* __restrict__ wsf, int* __restrict__ wsi,
    float* __restrict__ qty_out) {
  const int lane = threadIdx.x;
  float qv = 0.f, scale = 0.f;
  if (lane < NB) {
    float sum = wsf[OFF_SUMA + lane];
    float tg = (float)tgt[lane];
    qv = fabsf(sum - tg);
    scale = tg / fmaxf(sum, 1e-8f);
    wsf[OFF_SCALE + lane] = scale;
  }
  qv += __shfl_down(qv, 8);
  qv += __shfl_down(qv, 4);
  qv += __shfl_down(qv, 2);
  qv += __shfl_down(qv, 1);
  if (lane == 0) qty_out[0] = qv * (1.0f / NB);
  if (lane >= NB) return;

  const int b = lane;
  const float* ap = alpha + (long)b * NT;
  float* sa = wsf + OFF_SA + (long)b * NT;
  float* sr = wsf + OFF_SREM + (long)b * NT;
  float* so = wsf + OFF_SOVF + (long)b * NT;
  int* sp = wsi + OFF_SPOS + b * NT;
  int* cutF = wsi + OFF_CUTF + b * 17;
  int* cutT = wsi + OFF_CUTT + b * 17;
  cutF[0] = 0; cutT[0] = 0;

  float acc = 0.f;
  int F = 0, fidx = 0, lastFT = 0, nc = 1;
  for (int t = 0; t < NT; ++t) {
    if (nc < 16 && t == (nc << 8)) { cutF[nc] = F; cutT[nc] = lastFT; ++nc; }
    float a = ap[t] * scale;
    sa[t] = a;
    float na = acc + a;
    float rem = fmaxf(1.0f - acc, 0.0f);
    float ovf = fmaxf(a - rem, 0.0f);
    sr[t] = rem; so[t] = ovf;
    if (na >= 1.0f) {
      sp[t] = fidx;
      if (fidx < NT) ++fidx;
      ++F; lastFT = t; acc = ovf;
    } else {
      sp[t] = -1; acc = na;
    }
  }
  cutF[16] = F;
  wsi[OFF_FIDX + b] = fidx;
  wsi[OFF_TAIL + b] = (acc > 0.5f && fidx < NT) ? 1 : 0;
}

// ------------------------------------------------------------------
// Heavy pass: grid = B x 16 intervals x 4 d-chunks. Each block owns a
// contiguous run of fires -> plain, deterministic stores; enc read once.
__global__ __launch_bounds__(128) void k_cif(
    const float* __restrict__ enc, const int* __restrict__ lens,
    const float* __restrict__ wsf, const int* __restrict__ wsi,
    float* __restrict__ out) {
  const int bx = blockIdx.x;
  const int dc = bx & 3, iv = (bx >> 2) & 15, b = bx >> 6;
  const int d = (dc << 7) + threadIdx.x;
  const int* cutF = wsi + OFF_CUTF + b * 17;
  const int* cutT = wsi + OFF_CUTT + b * 17;
  const int Js = cutF[iv], Je = cutF[iv + 1];
  const bool last = (iv == 15);
  int tailb = 0, rowTail = 0;
  if (last) { tailb = wsi[OFF_TAIL + b]; rowTail = wsi[OFF_FIDX + b]; }
  if (Js == Je && (!last || !tailb)) return;

  const float* sa = wsf + OFF_SA + (long)b * NT;
  const float* sr = wsf + OFF_SREM + (long)b * NT;
  const float* so = wsf + OFF_SOVF + (long)b * NT;
  const int* sp = wsi + OFF_SPOS + b * NT;
  const float* eb = enc + (long)b * NT * ND;
  float* ob = out + (long)b * NT * ND;
  const int len = lens[b];

  int t = cutT[iv];
  int j = Js;
  float wsum = 0.f;
  if (Js > 0) {  // entry fire: only its overflow carries into this interval
    wsum = so[t] * eb[(long)t * ND + d];
    ++t;
  }
  for (; t < NT; ++t) {
    if (j >= Je) {
      if (!last) break;
      if (!tailb) break;
      if (t >= len) break;  // alpha==0 beyond len: tail partial complete
    }
    float h = eb[(long)t * ND + d];
    __builtin_prefetch(eb + (long)min(t + 8, NT - 1) * ND + d, 0, 1);
    int p = sp[t];
    if (p >= 0) {  // fire
      if (p < NT) ob[(long)p * ND + d] = wsum + sr[t] * h;
      wsum = so[t] * h;
      ++j;
    } else {
      wsum += sa[t] * h;
    }
  }
  if (last && tailb) {
    int r = rowTail < (NT - 1) ? rowTail : (NT - 1);
    ob[(long)r * ND + d] = wsum;
  }
}

// ------------------------------------------------------------------
extern "C" void kernel_launch(void* const* d_in, const int* in_sizes, int n_in,
                              void* d_out, int out_size, void* d_ws, size_t ws_size,
                              hipStream_t stream) {
  const float* enc   = (const float*)d_in[0];
  const int*   lens  = (const int*)d_in[1];
  const int*   tgt   = (const int*)d_in[2];
  const float* convw = (const float*)d_in[3];
  const float* gamma = (const float*)d_in[4];
  const float* beta  = (const float*)d_in[5];
  const float* projw = (const float*)d_in[6];
  const float* projb = (const float*)d_in[7];

  float* out   = (float*)d_out;
  float* alpha = out + (long)NB * NT * ND;
  float* qty   = alpha + (long)NB * NT;
  float* wsf   = (float*)d_ws;
  int*   wsi   = (int*)d_ws;

  long n = (long)out_size;
  k_zero<<<(unsigned)((n + 1023) / 1024), 256, 0, stream>>>(out, n);
  k_pre<<<1, 256, 0, stream>>>(gamma, beta, projw, wsf);
  k_alpha<<<NB * 256, 32, 0, stream>>>(enc, lens, convw, gamma, projw, wsf, projb, alpha);
  k_sum<<<NB, 256, 0, stream>>>(alpha, wsf);
  k_sched<<<1, 32, 0, stream>>>(alpha, lens, tgt, wsf, wsi, qty);
  k_cif<<<NB * 16 * 4, 128, 0, stream>>>(enc, lens, wsf, wsi, out);
}